// TreeMultiSimilarityLoss_40303973105926
// MI455X (gfx1250) — compile-verified
//
#include <hip/hip_runtime.h>
#include <hip/hip_bf16.h>
#include <math.h>

// ---------------------------------------------------------------------------
// TreeMultiSimilarityLoss on MI455X (gfx1250, wave32)
//
// Bandwidth-bound streaming kernel (512 MB @ 23.3 TB/s => ~22us floor).
// Per-row length-128 reductions are done on the matrix unit via the
// "ones-matmul" trick with V_WMMA_F32_16X16X4_F32 (f32-native, no precision
// loss): C = Ones(16x4) x B(4x16) accumulated over 32 K-chunks gives every
// lane its row's full sum in c[0] with no cross-lane extraction.
// ---------------------------------------------------------------------------

typedef __attribute__((ext_vector_type(2))) float v2f;
typedef __attribute__((ext_vector_type(8))) float v8f;

#define NBUCKETS 7
#define NBINS    (3 * NBUCKETS)   // pos[7], neg[7], cnt[7]

__global__ void tms_init(float* __restrict__ gAcc) {
    if (threadIdx.x < NBINS) gAcc[threadIdx.x] = 0.0f;
}

// blockDim.x = 320 (10 waves). Each wave owns `tilesPerWave` 16-row tiles.
__global__ __launch_bounds__(320) void tms_main(
    const float* __restrict__ q1, const float* __restrict__ q2,
    const int*   __restrict__ dists, float* __restrict__ gAcc,
    int numTiles, int tilesPerWave, int wavesTotal)
{
    __shared__ float sAcc[NBINS];
    const int tid = threadIdx.x;
    if (tid < NBINS) sAcc[tid] = 0.0f;
    __syncthreads();

    const int lane   = tid & 31;
    const int waveId = blockIdx.x * (blockDim.x >> 5) + (tid >> 5);
    const int r = lane & 15;   // row-in-tile == B-matrix column j
    const int h = lane >> 4;   // which K-pair of the 4-wide chunk this lane holds

    const v2f ones = {1.0f, 1.0f};   // A = Ones(16x4): layout-independent

    for (int it = 0; it < tilesPerWave; ++it) {
        const int tile = waveId + it * wavesTotal;   // uniform within the wave
        if (tile < numTiles) {                       // always true for exact split
            const size_t rowOff = (size_t)(tile * 16 + r) * 128 + 2 * h;
            const float* p1 = q1 + rowOff;
            const float* p2 = q2 + rowOff;

            v8f cDot = {}; v8f cN1 = {}; v8f cN2 = {};

            // 32 chunks of K=4 cover D=128. B layout (4x16 f32, 2 VGPRs):
            // lane l -> column j=l%16, V0:K=2*(l/16), V1:K=2*(l/16)+1.
            #pragma unroll 4
            for (int k = 0; k < 32; ++k) {
                const float2 a = *reinterpret_cast<const float2*>(p1 + k * 4);
                const float2 b = *reinterpret_cast<const float2*>(p2 + k * 4);
                v2f bd = {a.x * b.x, a.y * b.y};   // q1*q2 chunk
                v2f b1 = {a.x * a.x, a.y * a.y};   // q1*q1 chunk
                v2f b2 = {b.x * b.x, b.y * b.y};   // q2*q2 chunk
                cDot = __builtin_amdgcn_wmma_f32_16x16x4_f32(
                    false, ones, false, bd, (short)0, cDot, false, false);
                cN1  = __builtin_amdgcn_wmma_f32_16x16x4_f32(
                    false, ones, false, b1, (short)0, cN1, false, false);
                cN2  = __builtin_amdgcn_wmma_f32_16x16x4_f32(
                    false, ones, false, b2, (short)0, cN2, false, false);
            }

            // Every row of C is identical => lane<16 reads its row's sums
            // from c[0] (C layout: VGPR0 holds M=0, N=lane for lanes 0..15).
            if (lane < 16) {
                const float d  = cDot[0];
                const float s1 = cN1[0];
                const float s2 = cN2[0];
                const float denom = fmaxf(sqrtf(s1) * sqrtf(s2), 1e-8f);
                const float c = (d / denom + 1.0f) * 0.5f;
                // stable softplus: sp(x) = max(x,0) + log1p(exp(-|x|))
                const float x1 = 2.0f  * (0.5f - c);
                const float x2 = 50.0f * (c - 0.5f);
                const float pos = (fmaxf(x1, 0.0f) + log1pf(__expf(-fabsf(x1)))) * 0.5f;
                const float neg = (fmaxf(x2, 0.0f) + log1pf(__expf(-fabsf(x2)))) * 0.02f;
                const int bkt = dists[tile * 16 + lane];
                atomicAdd(&sAcc[bkt],                pos);
                atomicAdd(&sAcc[NBUCKETS + bkt],     neg);
                atomicAdd(&sAcc[2 * NBUCKETS + bkt], 1.0f);
            }
        }
    }

    __syncthreads();
    if (tid < NBINS) atomicAdd(&gAcc[tid], sAcc[tid]);
}

// One wave: loss = sum_i (6-i)*mean_pos[i] + i*mean_neg[i]
__global__ void tms_final(const float* __restrict__ gAcc, float* __restrict__ out) {
    const int i = threadIdx.x;
    float t = 0.0f;
    if (i < NBUCKETS) {
        const float cnt = gAcc[2 * NBUCKETS + i];
        t = ((float)(NBUCKETS - 1 - i) * gAcc[i] +
             (float)i * gAcc[NBUCKETS + i]) / cnt;
    }
    #pragma unroll
    for (int m = 16; m >= 1; m >>= 1) t += __shfl_xor(t, m, 32);
    if (i == 0) out[0] = t;
}

extern "C" void kernel_launch(void* const* d_in, const int* in_sizes, int n_in,
                              void* d_out, int out_size, void* d_ws, size_t ws_size,
                              hipStream_t stream) {
    const float* q1    = (const float*)d_in[0];
    const float* q2    = (const float*)d_in[1];
    const int*   dists = (const int*)d_in[2];
    float* out  = (float*)d_out;
    float* gAcc = (float*)d_ws;

    const int N        = in_sizes[2];       // 500000
    const int numTiles = N / 16;            // 31250 (N divisible by 16)

    const int blocks        = 625;
    const int wavesPerBlock = 10;           // 320 threads
    const int wavesTotal    = blocks * wavesPerBlock;          // 6250
    const int tilesPerWave  = (numTiles + wavesTotal - 1) / wavesTotal;  // 5, exact

    tms_init<<<1, 32, 0, stream>>>(gAcc);
    tms_main<<<blocks, wavesPerBlock * 32, 0, stream>>>(
        q1, q2, dists, gAcc, numTiles, tilesPerWave, wavesTotal);
    tms_final<<<1, 32, 0, stream>>>(gAcc, out);
}